// LDAMLossOLD_30880814858588
// MI455X (gfx1250) — compile-verified
//
#include <hip/hip_runtime.h>

// LDAM loss, two streaming passes, deterministic reductions.
// WMMA (v_wmma_f32_16x16x4_f32) used for exact f32 wave32 all-reduce.

typedef __attribute__((ext_vector_type(2))) float v2f;
typedef __attribute__((ext_vector_type(8))) float v8f;

#define BLOCK 256
#define C_CONST 1.0f
#define MAX_M 0.5f

__device__ __forceinline__ float fast_sigmoid(float x) {
    return 1.0f / (1.0f + __expf(-x));
}

// C / cc^0.25 = C * rsqrt(sqrt(cc)), cc >= 1
__device__ __forceinline__ float inv_cc4(float cc) {
    return C_CONST * rsqrtf(sqrtf(cc));
}

// Exact f32 all-reduce (sum) across one full wave32 via V_WMMA_F32_16X16X4_F32.
// A layout (16x4 f32): lanes 0-15 hold A[m][0],A[m][1]; lanes 16-31 hold
// A[m][2],A[m][3]. We put x in slot 0/2, zero in 1/3. B = all ones (4x16,
// layout independent), so D[m][n] = x[m] + x[m+16] for every n.
// D layout: vgpr r, lanes 0-15 -> M=r, lanes 16-31 -> M=r+8. Summing the 8
// D vgprs per lane gives the half-sums; one shfl_xor(16) completes it.
// Requires EXEC all ones for the wave (call from fully-active waves only).
__device__ __forceinline__ float wave_allreduce_sum_wmma(float x) {
    v2f a; a[0] = x;    a[1] = 0.0f;
    v2f b; b[0] = 1.0f; b[1] = 1.0f;
    v8f c = {};
    v8f d = __builtin_amdgcn_wmma_f32_16x16x4_f32(
        /*neg_a=*/false, a, /*neg_b=*/false, b,
        /*c_mod=*/(short)0, c, /*reuse_a=*/false, /*reuse_b=*/false);
    float s = d[0] + d[1] + d[2] + d[3] + d[4] + d[5] + d[6] + d[7];
    s += __shfl_xor(s, 16, 32);
    return s;
}

// Deterministic block (256-thread) sum. Result valid in threads 0..31.
__device__ __forceinline__ float block_sum(float v, float* lds) {
    const int t = threadIdx.x;
    lds[t] = v;
    __syncthreads();
    float total = 0.0f;
    if (t < 32) {   // wave 0 fully active -> EXEC all ones for WMMA
        float s = 0.0f;
#pragma unroll
        for (int i = 0; i < BLOCK / 32; ++i) s += lds[t + i * 32];
        total = wave_allreduce_sum_wmma(s);
    }
    __syncthreads();
    return total;
}

// Deterministic block max. Result valid in threads 0..31.
__device__ __forceinline__ float block_max(float v, float* lds) {
    const int t = threadIdx.x;
    lds[t] = v;
    __syncthreads();
    float m = 0.0f;
    if (t < 32) {
        m = lds[t];
#pragma unroll
        for (int i = 1; i < BLOCK / 32; ++i) m = fmaxf(m, lds[t + i * 32]);
#pragma unroll
        for (int off = 16; off >= 1; off >>= 1)
            m = fmaxf(m, __shfl_xor(m, off, 32));
    }
    __syncthreads();
    return m;
}

__global__ void k_init(unsigned int* gmax) {
    if (threadIdx.x < 2) gmax[threadIdx.x] = 0u;   // 0u == +0.0f bits
}

// Pass 1: per-row sums + global masked maxes of C/cc^0.25.
__global__ void __launch_bounds__(BLOCK)
k_rows(const float* __restrict__ logits, const float* __restrict__ labels,
       const float* __restrict__ cc,
       float* __restrict__ rowPos, float* __restrict__ rowNeg,
       float* __restrict__ rowCnt, unsigned int* __restrict__ gmax, int L) {
    __shared__ float lds[BLOCK];
    const int b = blockIdx.x;
    const int t = threadIdx.x;
    const float4* lg4 = (const float4*)(logits + (size_t)b * L);
    const float4* lb4 = (const float4*)(labels + (size_t)b * L);
    const float4* cc4 = (const float4*)cc;
    const int n4 = L >> 2;

    float sPos = 0.0f, sNeg = 0.0f, sCnt = 0.0f, mPos = 0.0f, mNeg = 0.0f;
    for (int i = t; i < n4; i += BLOCK) {
        float4 x = lg4[i];
        float4 l = lb4[i];
        float4 q = cc4[i];
#define DO_ELEM(f)                                                     \
        {                                                              \
            float p = fast_sigmoid(x.f);                               \
            float ic = inv_cc4(q.f);                                   \
            float pl = p * l.f;                                        \
            sPos += pl; sNeg += p - pl; sCnt += l.f;                   \
            if (l.f > 0.0f) mPos = fmaxf(mPos, ic);                    \
            else            mNeg = fmaxf(mNeg, ic);                    \
        }
        DO_ELEM(x) DO_ELEM(y) DO_ELEM(z) DO_ELEM(w)
#undef DO_ELEM
    }

    float tPos = block_sum(sPos, lds);
    float tNeg = block_sum(sNeg, lds);
    float tCnt = block_sum(sCnt, lds);
    float bPos = block_max(mPos, lds);
    float bNeg = block_max(mNeg, lds);
    if (t == 0) {
        rowPos[b] = tPos;
        rowNeg[b] = tNeg;
        rowCnt[b] = tCnt;
        // positive floats order as uints; max is order-independent => deterministic
        atomicMax(&gmax[0], __float_as_uint(bPos));
        atomicMax(&gmax[1], __float_as_uint(bNeg));
    }
}

// Reduce row stats -> scalars: sc[0]=total_pos sc[1]=total_neg(total_pos2)
// sc[2]=sumLabels sc[3]=sumInv sc[4]=scale1 sc[5]=scale2
__global__ void __launch_bounds__(BLOCK)
k_scalars(const float* __restrict__ rowPos, const float* __restrict__ rowNeg,
          const float* __restrict__ rowCnt, const unsigned int* __restrict__ gmax,
          float* __restrict__ sc, int B, int L) {
    __shared__ float lds[BLOCK];
    const int t = threadIdx.x;
    float sP = 0.0f, sN = 0.0f, sC = 0.0f;
    for (int i = t; i < B; i += BLOCK) {
        sP += rowPos[i];
        sN += rowNeg[i];
        sC += rowCnt[i];
    }
    sP = block_sum(sP, lds);
    sN = block_sum(sN, lds);
    sC = block_sum(sC, lds);
    if (t == 0) {
        sc[0] = sP;
        sc[1] = sN;
        sc[2] = sC;
        sc[3] = (float)B * (float)L - sC;
        sc[4] = MAX_M / __uint_as_float(gmax[0]);
        sc[5] = MAX_M / __uint_as_float(gmax[1]);
    }
}

// Pass 2: masked -log terms. part1 term = log(den1) - (p - d1) on positives.
// part2 term = -log(1 - num2/den2) = log(den2) - log(salt2 - p) on negatives,
// since den2 - num2 = salt2 - p.
__global__ void __launch_bounds__(BLOCK)
k_loss(const float* __restrict__ logits, const float* __restrict__ labels,
       const float* __restrict__ cc, const float* __restrict__ rowPos,
       const float* __restrict__ rowNeg, const float* __restrict__ sc,
       float* __restrict__ blockAcc1, float* __restrict__ blockAcc2, int L) {
    __shared__ float lds[BLOCK];
    const int b = blockIdx.x;
    const int t = threadIdx.x;
    const float total_pos = sc[0];
    const float total_neg = sc[1];
    const float scale1 = sc[4];
    const float scale2 = sc[5];
    const float sum_all1 = rowNeg[b] + total_pos;   // neg_row + total_pos
    const float salt2    = rowPos[b] + total_neg;   // neg_row2 + total_pos2

    const float4* lg4 = (const float4*)(logits + (size_t)b * L);
    const float4* lb4 = (const float4*)(labels + (size_t)b * L);
    const float4* cc4 = (const float4*)cc;
    const int n4 = L >> 2;

    float acc1 = 0.0f, acc2 = 0.0f;
    for (int i = t; i < n4; i += BLOCK) {
        float4 x = lg4[i];
        float4 l = lb4[i];
        float4 q = cc4[i];
#define DO_ELEM(f)                                                     \
        {                                                              \
            float p = fast_sigmoid(x.f);                               \
            float ic = inv_cc4(q.f);                                   \
            bool pos = l.f > 0.0f;                                     \
            float d = ic * (pos ? scale1 : scale2);                    \
            float num = __expf(p - d);                                 \
            float base = pos ? sum_all1 : salt2;                       \
            float den = num + base - p;                                \
            float lgden = __logf(den);                                 \
            if (pos) acc1 += lgden - (p - d);                          \
            else     acc2 += lgden - __logf(salt2 - p);                \
        }
        DO_ELEM(x) DO_ELEM(y) DO_ELEM(z) DO_ELEM(w)
#undef DO_ELEM
    }

    float a1 = block_sum(acc1, lds);
    float a2 = block_sum(acc2, lds);
    if (t == 0) {
        blockAcc1[b] = a1;
        blockAcc2[b] = a2;
    }
}

__global__ void __launch_bounds__(BLOCK)
k_final(const float* __restrict__ blockAcc1, const float* __restrict__ blockAcc2,
        const float* __restrict__ sc, float* __restrict__ out, int B) {
    __shared__ float lds[BLOCK];
    const int t = threadIdx.x;
    float a1 = 0.0f, a2 = 0.0f;
    for (int i = t; i < B; i += BLOCK) {
        a1 += blockAcc1[i];
        a2 += blockAcc2[i];
    }
    a1 = block_sum(a1, lds);
    a2 = block_sum(a2, lds);
    if (t == 0) out[0] = a1 / sc[2] + a2 / sc[3];
}

extern "C" void kernel_launch(void* const* d_in, const int* in_sizes, int n_in,
                              void* d_out, int out_size, void* d_ws, size_t ws_size,
                              hipStream_t stream) {
    const float* logits = (const float*)d_in[0];
    const float* labels = (const float*)d_in[1];
    const float* cc     = (const float*)d_in[2];
    const int L = in_sizes[2];
    const int B = in_sizes[0] / L;

    float* ws = (float*)d_ws;
    float* rowPos    = ws;
    float* rowNeg    = ws + (size_t)B;
    float* rowCnt    = ws + (size_t)2 * B;
    float* blockAcc1 = ws + (size_t)3 * B;
    float* blockAcc2 = ws + (size_t)4 * B;
    unsigned int* gmax = (unsigned int*)(ws + (size_t)5 * B);
    float* sc        = ws + (size_t)5 * B + 2;

    k_init<<<1, 32, 0, stream>>>(gmax);
    k_rows<<<B, BLOCK, 0, stream>>>(logits, labels, cc, rowPos, rowNeg, rowCnt,
                                    gmax, L);
    k_scalars<<<1, BLOCK, 0, stream>>>(rowPos, rowNeg, rowCnt, gmax, sc, B, L);
    k_loss<<<B, BLOCK, 0, stream>>>(logits, labels, cc, rowPos, rowNeg, sc,
                                    blockAcc1, blockAcc2, L);
    k_final<<<1, BLOCK, 0, stream>>>(blockAcc1, blockAcc2, sc, (float*)d_out, B);
}